// Model_84885733638152
// MI455X (gfx1250) — compile-verified
//
#include <hip/hip_runtime.h>
#include <hip/hip_bf16.h>

typedef __attribute__((ext_vector_type(2))) float v2f;
typedef __attribute__((ext_vector_type(8))) float v8f;

#define NUM_USERS  100000
#define NUM_MOVIES 50000
#define NUM_EDGES  4000000
#define USER_FEAT  24
#define MOVIE_FEAT 404
#define HID        64
#define GENRES     3

// ---------------------------------------------------------------------------
// Degree count: deg[idx[e]] += 1.0f
// ---------------------------------------------------------------------------
__global__ __launch_bounds__(256) void degree_kernel(const int* __restrict__ idx,
                                                     float* __restrict__ deg, int E) {
    int e = blockIdx.x * blockDim.x + threadIdx.x;
    if (e < E) atomicAdd(&deg[idx[e]], 1.0f);
}

// ---------------------------------------------------------------------------
// Edge scatter-sum: acc[dst[e]] += X[src[e]]  (64 channels, 4 per thread)
// 16 consecutive threads cooperate on one edge -> fully coalesced 256B row.
// ---------------------------------------------------------------------------
__global__ __launch_bounds__(256) void scatter_kernel(const float* __restrict__ X,
                                                      const int* __restrict__ src,
                                                      const int* __restrict__ dst,
                                                      float* __restrict__ acc, int E) {
    long long t = (long long)blockIdx.x * blockDim.x + threadIdx.x;
    int e = (int)(t >> 4);
    int c = ((int)t & 15) * 4;
    if (e >= E) return;
    int s = src[e];
    int d = dst[e];
    const float4 v = *(const float4*)(X + (size_t)s * HID + c);
    float* o = acc + (size_t)d * HID + c;
    atomicAdd(o + 0, v.x);
    atomicAdd(o + 1, v.y);
    atomicAdd(o + 2, v.z);
    atomicAdd(o + 3, v.w);
}

// ---------------------------------------------------------------------------
// Fused WMMA GEMM, N fixed to 64:
//   C[M,64] = (A1 * rowscale) @ W1^T  (+ A2 @ W2^T)  (+ bias)  (relu?)
// rowscale = 1/max(deg1[row],1) when deg1 != nullptr (fused mean-aggregation).
// W layout is [64, K] row-major (PyTorch Linear convention), so B = W^T.
// One wave -> 16x64 strip via 4x V_WMMA_F32_16X16X4_F32 per K-step.
// ---------------------------------------------------------------------------
__global__ __launch_bounds__(256)
void sage_gemm64(const float* __restrict__ A1, const float* __restrict__ deg1,
                 const float* __restrict__ W1, int K1,
                 const float* __restrict__ A2, const float* __restrict__ W2, int K2,
                 const float* __restrict__ bias, float* __restrict__ C,
                 int M, int relu) {
    const int lane = threadIdx.x & 31;
    const int wave = threadIdx.x >> 5;
    const int half = lane >> 4;   // lane-half selects K = j + 2*half inside 4-wide K step
    const int r    = lane & 15;   // A: row within strip; B/C: column within 16-col tile
    const int row0 = (blockIdx.x * 8 + wave) * 16;
    if (row0 >= M) return;        // wave-uniform: EXEC stays all-ones for WMMA

    v8f c[4] = {};                // 4 col-tiles of 16 -> N = 64

    float scale = 1.0f;
    if (deg1) scale = 1.0f / fmaxf(deg1[row0 + r], 1.0f);

    { // first operand pair
        const float* Arow = A1 + (size_t)(row0 + r) * K1;
        for (int kk = 0; kk < K1; kk += 4) {
            const int k = kk + 2 * half;
            v2f a;
            a.x = Arow[k] * scale;      // VGPR0: K = kk + 2h
            a.y = Arow[k + 1] * scale;  // VGPR1: K = kk + 2h + 1
#pragma unroll
            for (int t = 0; t < 4; ++t) {
                const float* Wrow = W1 + (size_t)(t * 16 + r) * K1;
                v2f b;
                b.x = Wrow[k];          // B[k][n] = W[n][k]
                b.y = Wrow[k + 1];
                c[t] = __builtin_amdgcn_wmma_f32_16x16x4_f32(
                    false, a, false, b, (short)0, c[t], false, false);
            }
        }
    }
    if (A2) { // second operand pair (lin_r term of SAGEConv)
        const float* Arow = A2 + (size_t)(row0 + r) * K2;
        for (int kk = 0; kk < K2; kk += 4) {
            const int k = kk + 2 * half;
            v2f a;
            a.x = Arow[k];
            a.y = Arow[k + 1];
#pragma unroll
            for (int t = 0; t < 4; ++t) {
                const float* Wrow = W2 + (size_t)(t * 16 + r) * K2;
                v2f b;
                b.x = Wrow[k];
                b.y = Wrow[k + 1];
                c[t] = __builtin_amdgcn_wmma_f32_16x16x4_f32(
                    false, a, false, b, (short)0, c[t], false, false);
            }
        }
    }

    // C layout: element i of v8f at lane -> row = row0 + i + 8*half, col = t*16 + r
#pragma unroll
    for (int t = 0; t < 4; ++t) {
        const float bv = bias ? bias[t * 16 + r] : 0.0f;
#pragma unroll
        for (int i = 0; i < 8; ++i) {
            float v = c[t][i] + bv;
            if (relu) v = fmaxf(v, 0.0f);
            C[(size_t)(row0 + i + 8 * half) * HID + t * 16 + r] = v;
        }
    }
}

// ---------------------------------------------------------------------------
// Final head (N = GENRES = 3): out = (acc2/deg) @ Wl2^T + bl2 + m1 @ Wr2^T
// Tiny N -> plain VALU kernel, one movie per thread.
// ---------------------------------------------------------------------------
__global__ __launch_bounds__(256)
void head_kernel(const float* __restrict__ acc2, const float* __restrict__ degm,
                 const float* __restrict__ m1, const float* __restrict__ Wl2,
                 const float* __restrict__ bl2, const float* __restrict__ Wr2,
                 float* __restrict__ out, int M) {
    int m = blockIdx.x * blockDim.x + threadIdx.x;
    if (m >= M) return;
    const float inv = 1.0f / fmaxf(degm[m], 1.0f);
    const float* a = acc2 + (size_t)m * HID;
    const float* x = m1 + (size_t)m * HID;
    float acc[GENRES];
#pragma unroll
    for (int g = 0; g < GENRES; ++g) acc[g] = bl2[g];
    for (int k = 0; k < HID; ++k) {
        const float av = a[k] * inv;
        const float xv = x[k];
#pragma unroll
        for (int g = 0; g < GENRES; ++g)
            acc[g] += av * Wl2[g * HID + k] + xv * Wr2[g * HID + k];
    }
#pragma unroll
    for (int g = 0; g < GENRES; ++g) out[(size_t)m * GENRES + g] = acc[g];
}

// ---------------------------------------------------------------------------
extern "C" void kernel_launch(void* const* d_in, const int* in_sizes, int n_in,
                              void* d_out, int out_size, void* d_ws, size_t ws_size,
                              hipStream_t stream) {
    (void)in_sizes; (void)n_in; (void)out_size; (void)ws_size;

    const float* x_user  = (const float*)d_in[0];
    const float* x_movie = (const float*)d_in[1];
    const int*   e_src   = (const int*)d_in[2];
    const int*   e_dst   = (const int*)d_in[3];
    const float* W_user  = (const float*)d_in[4];
    const float* b_user  = (const float*)d_in[5];
    const float* W_movie = (const float*)d_in[6];
    const float* b_movie = (const float*)d_in[7];
    const float* Wl1_um  = (const float*)d_in[8];
    const float* bl1_um  = (const float*)d_in[9];
    const float* Wr1_um  = (const float*)d_in[10];
    const float* Wl1_mu  = (const float*)d_in[11];
    const float* bl1_mu  = (const float*)d_in[12];
    const float* Wr1_mu  = (const float*)d_in[13];
    const float* Wl2_um  = (const float*)d_in[14];
    const float* bl2_um  = (const float*)d_in[15];
    const float* Wr2_um  = (const float*)d_in[16];
    float* out = (float*)d_out;

    // Workspace layout (floats). Zeroed region [acc_um .. deg_movie] is contiguous.
    float* ws = (float*)d_ws;
    float* acc_um    = ws;                          // 50000*64  = 3,200,000
    float* acc_mu    = acc_um + 3200000;            // 100000*64 = 6,400,000
    float* acc2      = acc_mu + 6400000;            // 50000*64  = 3,200,000
    float* deg_user  = acc2 + 3200000;              // 100,000
    float* deg_movie = deg_user + 100000;           // 50,000
    float* hu        = deg_movie + 50000;           // 100000*64 = 6,400,000
    float* hm        = hu + 6400000;                // 50000*64  = 3,200,000
    float* m1        = hm + 3200000;                // 50000*64  = 3,200,000
    float* u1        = m1 + 3200000;                // 100000*64 = 6,400,000

    const size_t zero_bytes = (size_t)(3200000 + 6400000 + 3200000 + 100000 + 50000) * 4;
    hipMemsetAsync(d_ws, 0, zero_bytes, stream);

    const int E = NUM_EDGES;
    // Degrees: movie in-degree (edge_dst), user degree (edge_src)
    degree_kernel<<<(E + 255) / 256, 256, 0, stream>>>(e_dst, deg_movie, E);
    degree_kernel<<<(E + 255) / 256, 256, 0, stream>>>(e_src, deg_user, E);

    // Input projections (WMMA f32):  hu = x_user @ W_user^T + b_user ; hm likewise
    const int gridU = (NUM_USERS / 16 + 7) / 8;    // 8 waves (16-row strips) per block
    const int gridM = (NUM_MOVIES / 16 + 7) / 8;
    sage_gemm64<<<gridU, 256, 0, stream>>>(x_user, nullptr, W_user, USER_FEAT,
                                           nullptr, nullptr, 0, b_user, hu,
                                           NUM_USERS, 0);
    sage_gemm64<<<gridM, 256, 0, stream>>>(x_movie, nullptr, W_movie, MOVIE_FEAT,
                                           nullptr, nullptr, 0, b_movie, hm,
                                           NUM_MOVIES, 0);

    const long long sthreads = (long long)E * 16;
    const int sblocks = (int)((sthreads + 255) / 256);

    // Layer 1, movie side: mean over hu[src] grouped by dst, then fused SAGE + ReLU
    scatter_kernel<<<sblocks, 256, 0, stream>>>(hu, e_src, e_dst, acc_um, E);
    sage_gemm64<<<gridM, 256, 0, stream>>>(acc_um, deg_movie, Wl1_um, HID,
                                           hm, Wr1_um, HID, bl1_um, m1,
                                           NUM_MOVIES, 1);

    // Layer 1, user side: mean over hm[dst] grouped by src
    scatter_kernel<<<sblocks, 256, 0, stream>>>(hm, e_dst, e_src, acc_mu, E);
    sage_gemm64<<<gridU, 256, 0, stream>>>(acc_mu, deg_user, Wl1_mu, HID,
                                           hu, Wr1_mu, HID, bl1_mu, u1,
                                           NUM_USERS, 1);

    // Layer 2: mean over u1[src] grouped by dst, tiny N=3 head
    scatter_kernel<<<sblocks, 256, 0, stream>>>(u1, e_src, e_dst, acc2, E);
    head_kernel<<<(NUM_MOVIES + 255) / 256, 256, 0, stream>>>(
        acc2, deg_movie, m1, Wl2_um, bl2_um, Wr2_um, out, NUM_MOVIES);
}